// DrugGCN_71244917506340
// MI455X (gfx1250) — compile-verified
//
#include <hip/hip_runtime.h>

typedef __attribute__((ext_vector_type(2))) float v2f;
typedef __attribute__((ext_vector_type(8))) float v8f;

#define GD 64  // feature dim

// ---------------------------------------------------------------------------
// Degree / normalization kernels
// ---------------------------------------------------------------------------
__global__ void k_init_deg(float* __restrict__ deg, int n) {
    int i = blockIdx.x * blockDim.x + threadIdx.x;
    if (i < n) deg[i] = 1.0f;                       // self-loop
}

__global__ void k_count_deg(const int* __restrict__ dst, float* __restrict__ deg, int e) {
    int i = blockIdx.x * blockDim.x + threadIdx.x;
    if (i < e) atomicAdd(&deg[dst[i]], 1.0f);
}

__global__ void k_dinv(const float* __restrict__ deg, float* __restrict__ dinv, int n) {
    int i = blockIdx.x * blockDim.x + threadIdx.x;
    if (i < n) dinv[i] = rsqrtf(fmaxf(deg[i], 1e-12f));
}

__global__ void k_norm(const int* __restrict__ src, const int* __restrict__ dst,
                       const float* __restrict__ dinv, float* __restrict__ norm, int e) {
    int i = blockIdx.x * blockDim.x + threadIdx.x;
    if (i < e) norm[i] = dinv[src[i]] * dinv[dst[i]];
}

// ---------------------------------------------------------------------------
// WMMA fp32 GEMM:  out[N,64] = act( H[N,64] @ W[64,64] (+ bias) )
// Template flags compile away all epilogue/prologue selects.
// 256 threads = 8 waves; each wave computes a 16-row stripe x 64 cols
// via 4 accumulators of V_WMMA_F32_16X16X4_F32 (K-loop of 16).
// W staged in LDS PRE-PAIRED: sWp[m][c] = {W[2m][c], W[2m+1][c]} so each
// B fragment is a single aligned ds_load_b64 straight into a VGPR pair.
// ---------------------------------------------------------------------------
template <bool RELU_IN, bool ADD_BIAS, bool RELU_OUT>
__global__ void __launch_bounds__(256)
k_gemm64(const float* __restrict__ H, const float* __restrict__ W,
         const float* __restrict__ bias, float* __restrict__ out, int n)
{
    __shared__ v2f   sWp[(GD / 2) * GD];   // 32 K-pairs x 64 cols, 16 KB
    __shared__ float sB[GD];

    const int t = threadIdx.x;
    for (int i = t; i < (GD / 2) * GD; i += 256) {
        const int m = i >> 6;          // K-pair index (rows 2m, 2m+1)
        const int c = i & 63;          // column
        v2f p;
        p.x = W[(2 * m) * GD + c];
        p.y = W[(2 * m + 1) * GD + c];
        sWp[i] = p;
    }
    if (ADD_BIAS && t < GD) sB[t] = bias[t];
    __syncthreads();

    const int wave = t >> 5;          // 0..7
    const int lane = t & 31;
    const int r    = lane & 15;       // 0..15
    const int hi   = lane >> 4;       // 0 or 1
    const int rowBase = blockIdx.x * 128 + wave * 16;
    if (rowBase >= n) return;         // wave-uniform: EXEC stays all-ones

    v8f acc0 = {}, acc1 = {}, acc2 = {}, acc3 = {};
    const float* aRow = H + (size_t)(rowBase + r) * GD;

    #pragma unroll
    for (int k = 0; k < 16; ++k) {
        const int ka = 4 * k + 2 * hi;            // K index of this lane's A pair
        v2f a = *(const v2f*)(aRow + ka);         // contiguous 8B load
        if (RELU_IN) { a.x = fmaxf(a.x, 0.0f); a.y = fmaxf(a.y, 0.0f); }

        const int mB = 2 * k + hi;                // K-pair row in sWp
        const v2f* wp = sWp + (size_t)mB * GD + r;
        const v2f b0 = wp[0];
        const v2f b1 = wp[16];
        const v2f b2 = wp[32];
        const v2f b3 = wp[48];

        acc0 = __builtin_amdgcn_wmma_f32_16x16x4_f32(false, a, false, b0, (short)0, acc0, false, false);
        acc1 = __builtin_amdgcn_wmma_f32_16x16x4_f32(false, a, false, b1, (short)0, acc1, false, false);
        acc2 = __builtin_amdgcn_wmma_f32_16x16x4_f32(false, a, false, b2, (short)0, acc2, false, false);
        acc3 = __builtin_amdgcn_wmma_f32_16x16x4_f32(false, a, false, b3, (short)0, acc3, false, false);
    }

    // C/D layout: VGPR v -> row rowBase + v + 8*hi, col = tile*16 + r
    #pragma unroll
    for (int v = 0; v < 8; ++v) {
        const int row = rowBase + v + 8 * hi;
        float* o = out + (size_t)row * GD + r;
        float x0 = acc0[v], x1 = acc1[v], x2 = acc2[v], x3 = acc3[v];
        if (ADD_BIAS) { x0 += sB[r]; x1 += sB[16 + r]; x2 += sB[32 + r]; x3 += sB[48 + r]; }
        if (RELU_OUT) {
            x0 = fmaxf(x0, 0.0f); x1 = fmaxf(x1, 0.0f);
            x2 = fmaxf(x2, 0.0f); x3 = fmaxf(x3, 0.0f);
        }
        o[0] = x0; o[16] = x1; o[32] = x2; o[48] = x3;
    }
}

// ---------------------------------------------------------------------------
// Aggregation init: out[i,:] = b + t[i,:] * dinv[i]^2   (self-loop term, no atomics)
// ---------------------------------------------------------------------------
__global__ void k_self_bias(const float* __restrict__ t, const float* __restrict__ dinv,
                            const float* __restrict__ b, float* __restrict__ out, int n)
{
    int tid = blockIdx.x * blockDim.x + threadIdx.x;
    if (tid >= n * 16) return;
    const int i = tid >> 4;
    const int c = tid & 15;
    const float s = dinv[i] * dinv[i];
    const float4 v  = *(const float4*)(t + (size_t)i * GD + c * 4);
    const float4 bb = *(const float4*)(b + c * 4);
    float4 o;
    o.x = bb.x + v.x * s;
    o.y = bb.y + v.y * s;
    o.z = bb.z + v.z * s;
    o.w = bb.w + v.w * s;
    *(float4*)(out + (size_t)i * GD + c * 4) = o;
}

// ---------------------------------------------------------------------------
// Edge scatter: out[dst,:] += t[src,:] * norm[e]
// 16 threads per edge, float4 gather + 4 f32 atomics (resolve in L2).
// ---------------------------------------------------------------------------
__global__ void k_scatter(const float* __restrict__ t, const int* __restrict__ src,
                          const int* __restrict__ dst, const float* __restrict__ norm,
                          float* __restrict__ out, int e)
{
    int tid = blockIdx.x * blockDim.x + threadIdx.x;
    if (tid >= e * 16) return;
    const int ei = tid >> 4;
    const int c  = tid & 15;
    const int s  = src[ei];
    const int d  = dst[ei];
    const float w = norm[ei];
    const float4 v = *(const float4*)(t + (size_t)s * GD + c * 4);
    float* o = out + (size_t)d * GD + c * 4;
    atomicAdd(o + 0, v.x * w);
    atomicAdd(o + 1, v.y * w);
    atomicAdd(o + 2, v.z * w);
    atomicAdd(o + 3, v.w * w);
}

// ---------------------------------------------------------------------------
extern "C" void kernel_launch(void* const* d_in, const int* in_sizes, int n_in,
                              void* d_out, int out_size, void* d_ws, size_t ws_size,
                              hipStream_t stream)
{
    const float* x    = (const float*)d_in[0];
    const int*   ei   = (const int*)d_in[1];
    const float* W_in = (const float*)d_in[2];
    const float* b_in = (const float*)d_in[3];
    const float* W1   = (const float*)d_in[4];
    const float* b1   = (const float*)d_in[5];
    const float* W2   = (const float*)d_in[6];
    const float* b2   = (const float*)d_in[7];
    const float* W3   = (const float*)d_in[8];
    const float* b3   = (const float*)d_in[9];
    float* out = (float*)d_out;

    const int N = in_sizes[0] / GD;      // 50000
    const int E = in_sizes[1] / 2;       // 1.6M
    const int* src = ei;
    const int* dst = ei + E;

    // workspace layout (all fp32)
    float* bufA = (float*)d_ws;                 // N*64
    float* bufB = bufA + (size_t)N * GD;        // N*64
    float* deg  = bufB + (size_t)N * GD;        // N
    float* dinv = deg  + N;                     // N
    float* norm = dinv + N;                     // E

    const int TB = 256;
    const int gN  = (N + TB - 1) / TB;
    const int gE  = (E + TB - 1) / TB;
    const int gN16 = (N * 16 + TB - 1) / TB;
    const int gE16 = (int)(((long long)E * 16 + TB - 1) / TB);
    const int gGemm = (N + 127) / 128;

    // normalization (computed once, reused by all 3 conv layers)
    k_init_deg <<<gN, TB, 0, stream>>>(deg, N);
    k_count_deg<<<gE, TB, 0, stream>>>(dst, deg, E);
    k_dinv     <<<gN, TB, 0, stream>>>(deg, dinv, N);
    k_norm     <<<gE, TB, 0, stream>>>(src, dst, dinv, norm, E);

    // layer 0: h = relu(x @ W_in + b_in)            -> bufA
    k_gemm64<false, true, true><<<gGemm, TB, 0, stream>>>(x, W_in, b_in, bufA, N);

    // conv1: t = h @ W1; agg = scatter(t*norm)+b1   -> bufA (relu deferred to next gemm)
    k_gemm64<false, false, false><<<gGemm, TB, 0, stream>>>(bufA, W1, nullptr, bufB, N);
    k_self_bias<<<gN16, TB, 0, stream>>>(bufB, dinv, b1, bufA, N);
    k_scatter <<<gE16, TB, 0, stream>>>(bufB, src, dst, norm, bufA, E);

    // conv2: t = relu(agg) @ W2; aggregate          -> bufA
    k_gemm64<true, false, false><<<gGemm, TB, 0, stream>>>(bufA, W2, nullptr, bufB, N);
    k_self_bias<<<gN16, TB, 0, stream>>>(bufB, dinv, b2, bufA, N);
    k_scatter <<<gE16, TB, 0, stream>>>(bufB, src, dst, norm, bufA, E);

    // conv3 (no trailing relu): result straight into d_out
    k_gemm64<true, false, false><<<gGemm, TB, 0, stream>>>(bufA, W3, nullptr, bufB, N);
    k_self_bias<<<gN16, TB, 0, stream>>>(bufB, dinv, b3, out, N);
    k_scatter <<<gE16, TB, 0, stream>>>(bufB, src, dst, norm, out, E);
}